// LengthRegulator_72748156059958
// MI455X (gfx1250) — compile-verified
//
#include <hip/hip_runtime.h>
#include <hip/hip_bf16.h>
#include <stdint.h>

// Problem constants (from reference)
#define B_DIM 16
#define C_DIM 512
#define T_DIM 512
#define MAXF  4096
#define CPG   8          // channels per gather block
#define OUT_ELEMS   (B_DIM * C_DIM * MAXF)   // 33,554,432
#define PITCH_ELEMS (B_DIM * MAXF)           // 65,536

// ---- gfx1250 async global->LDS staging (guarded; falls back to vector LDS copy) ----
#if defined(__has_builtin)
#  if __has_builtin(__builtin_amdgcn_global_load_async_to_lds_b128)
#    define HAS_ASYNC_LDS 1
#  endif
#  if __has_builtin(__builtin_amdgcn_s_wait_asynccnt)
#    define HAS_WAIT_ASYNC_BUILTIN 1
#  endif
#endif

#ifdef HAS_ASYNC_LDS
// Builtin signature (from clang diagnostic): param0 = v4i in addrspace(1),
// param1 = v4i in addrspace(3), then constant-int offset and cpol.
typedef int v4i __attribute__((vector_size(16)));
typedef __attribute__((address_space(1))) v4i g_v4i;
typedef __attribute__((address_space(3))) v4i l_v4i;

__device__ __forceinline__ void async_copy16(const void* g, void* l) {
  __builtin_amdgcn_global_load_async_to_lds_b128(
      (g_v4i*)(void*)g, (l_v4i*)l, /*offset=*/0, /*cpol=*/0);
}
__device__ __forceinline__ void wait_async0() {
#ifdef HAS_WAIT_ASYNC_BUILTIN
  __builtin_amdgcn_s_wait_asynccnt(0);
#else
  asm volatile("s_wait_asynccnt 0" ::: "memory");
#endif
}
#endif

// -----------------------------------------------------------------------------
// Kernel 1: per-batch prefix sum of ds, searchsorted -> idx (-1 when masked),
// pitch output, total output.  16 blocks x 512 threads.
// -----------------------------------------------------------------------------
__global__ __launch_bounds__(T_DIM) void lr_setup(
    const float* __restrict__ notepitch,   // (B, T)
    const int*   __restrict__ ds,          // (B, T)
    int*         __restrict__ idx_ws,      // (B, MAXF)  workspace
    float*       __restrict__ pitch_out,   // (B, MAXF)
    int*         __restrict__ total_out)   // (B,)
{
  __shared__ int cum[T_DIM];
  const int t = threadIdx.x;
  const int b = blockIdx.x;

  cum[t] = ds[b * T_DIM + t];
  __syncthreads();

  // Hillis-Steele inclusive scan over 512 elements
  for (int off = 1; off < T_DIM; off <<= 1) {
    int v = cum[t];
    int a = (t >= off) ? cum[t - off] : 0;
    __syncthreads();
    cum[t] = v + a;
    __syncthreads();
  }

  int total = cum[T_DIM - 1];
  if (total == 0) {            // uniform across block: ds >= 0 so sum==0 => all zero
    __syncthreads();
    cum[t] = t + 1;            // ds replaced by ones -> cumsum = t+1
    total = T_DIM;
    __syncthreads();
  }
  if (t == 0) total_out[b] = total;

  // searchsorted(cum, pos, side='right'): first i with cum[i] > pos, clip to T-1
  for (int pos = t; pos < MAXF; pos += T_DIM) {
    int l = 0, r = T_DIM;
    while (l < r) {
      int m = (l + r) >> 1;
      if (cum[m] <= pos) l = m + 1; else r = m;
    }
    int i = (l < T_DIM - 1) ? l : (T_DIM - 1);
    bool valid = pos < total;
    idx_ws[b * MAXF + pos]    = valid ? i : -1;
    pitch_out[b * MAXF + pos] = valid ? notepitch[b * T_DIM + i] : 0.0f;
  }
}

// -----------------------------------------------------------------------------
// Kernel 2: gather xs rows through LDS.  Block = (batch, group of 8 channels).
// grid = 16 * (512/8) = 1024 blocks, 256 threads.
// LDS: 16KB idx row + 16KB of 8 xs rows = 32KB (WGP has 320KB).
// -----------------------------------------------------------------------------
__global__ __launch_bounds__(256) void lr_gather(
    const float* __restrict__ xs,       // (B, C, T)
    const int*   __restrict__ idx_ws,   // (B, MAXF)
    float*       __restrict__ out)      // (B, C, MAXF)
{
  __shared__ __align__(16) int   sidx[MAXF];        // 4096 ints
  __shared__ __align__(16) float sxs[CPG * T_DIM];  // 8 * 512 floats

  const int t  = threadIdx.x;
  const int b  = blockIdx.x >> 6;          // / (C_DIM/CPG)
  const int c0 = (blockIdx.x & 63) * CPG;  // first channel of this group

  const int*   gidx = idx_ws + b * MAXF;
  const float* gxs  = xs + ((size_t)(b * C_DIM + c0)) * T_DIM;  // 8 contiguous rows

  // Stage: 1024 x 16B chunks each for idx and xs (4 iterations per thread)
#ifdef HAS_ASYNC_LDS
  for (int i = t; i < 1024; i += 256) {
    async_copy16(gidx + i * 4, sidx + i * 4);
    async_copy16(gxs  + i * 4, sxs  + i * 4);
  }
  wait_async0();
#else
  for (int i = t; i < 1024; i += 256) {
    ((int4*)sidx)[i]  = ((const int4*)gidx)[i];
    ((float4*)sxs)[i] = ((const float4*)gxs)[i];
  }
#endif
  __syncthreads();

  // Emit: per channel, 1024 float4 coalesced stores; gather hits LDS.
  for (int ch = 0; ch < CPG; ++ch) {
    float* orow = out + ((size_t)(b * C_DIM + c0 + ch)) * MAXF;
    const float* srow = sxs + ch * T_DIM;
    for (int v = t; v < 1024; v += 256) {
      int4 i4 = ((const int4*)sidx)[v];
      float4 o;
      o.x = (i4.x >= 0) ? srow[i4.x] : 0.0f;
      o.y = (i4.y >= 0) ? srow[i4.y] : 0.0f;
      o.z = (i4.z >= 0) ? srow[i4.z] : 0.0f;
      o.w = (i4.w >= 0) ? srow[i4.w] : 0.0f;
      ((float4*)orow)[v] = o;
    }
  }
}

// -----------------------------------------------------------------------------
extern "C" void kernel_launch(void* const* d_in, const int* in_sizes, int n_in,
                              void* d_out, int out_size, void* d_ws, size_t ws_size,
                              hipStream_t stream) {
  const float* xs        = (const float*)d_in[0];  // (16,512,512)
  const float* notepitch = (const float*)d_in[1];  // (16,512)
  const int*   ds        = (const int*)  d_in[2];  // (16,512)
  // d_in[3] = x_lengths: unused by the reference computation.

  float* out_buf   = (float*)d_out;                   // (16,512,4096)
  float* pitch_buf = out_buf + OUT_ELEMS;             // (16,4096)
  int*   total_buf = (int*)(pitch_buf + PITCH_ELEMS); // (16,) int32 bits

  int* idx_ws = (int*)d_ws;                           // (16,4096) = 256KB scratch

  lr_setup<<<dim3(B_DIM), dim3(T_DIM), 0, stream>>>(
      notepitch, ds, idx_ws, pitch_buf, total_buf);

  lr_gather<<<dim3(B_DIM * (C_DIM / CPG)), dim3(256), 0, stream>>>(
      xs, idx_ws, out_buf);
}